// neigh_26834955665989
// MI455X (gfx1250) — compile-verified
//
#include <hip/hip_runtime.h>

typedef __attribute__((ext_vector_type(2))) float v2f;
typedef __attribute__((ext_vector_type(8))) float v8f;

#define BB 32
#define NNODES 1024
#define FD 64
#define KC 64
#define LDSS 66  // padded LDS row stride (floats): 8B-aligned rows, avoids bank conflicts

// ---------------------------------------------------------------------------
// Y[b,n,:] = X[b,n,:] @ W + bias          (memory-bound; W staged in LDS)
// ---------------------------------------------------------------------------
__global__ __launch_bounds__(256) void linear_kernel(
    const float* __restrict__ X, const float* __restrict__ W,
    const float* __restrict__ bias, float* __restrict__ Y)
{
    __shared__ float sW[FD * FD];
    __shared__ float sX[4][FD];

    const int tid = threadIdx.x;
    for (int t = tid; t < FD * FD; t += 256) sW[t] = W[t];

    const int r0 = blockIdx.x * 4;
    const int rr = tid >> 6;    // 0..3 row within block
    const int f  = tid & 63;    // feature
    sX[rr][f] = X[(size_t)(r0 + rr) * FD + f];
    __syncthreads();

    float acc = bias[f];
#pragma unroll
    for (int k = 0; k < FD; ++k)
        acc += sX[rr][k] * sW[k * FD + f];
    Y[(size_t)(r0 + rr) * FD + f] = acc;
}

// ---------------------------------------------------------------------------
// H[b,i,f] = mask(i) * sum_j adj[b,i,j] * Y[b,j,f]   via V_WMMA_F32_16X16X4_F32
// Block: 128 threads = 4 waves; block computes 16 rows x 64 features.
// Wave w owns the 16x16 tile for features [16w, 16w+16).
// ---------------------------------------------------------------------------
__global__ __launch_bounds__(128) void agg_wmma_kernel(
    const float* __restrict__ adj, const float* __restrict__ Y,
    const int* __restrict__ num_nodes, float* __restrict__ H)
{
    __shared__ float ldsA[16 * LDSS];   // A tile: adj rows [16][KC]
    __shared__ float ldsY[FD * LDSS];   // Y tile transposed: [feature][k]

    const int i0   = blockIdx.x * 16;   // row tile
    const int b    = blockIdx.y;        // batch
    const int tid  = threadIdx.x;
    const int wave = tid >> 5;
    const int lane = tid & 31;
    const int half = lane >> 4;         // 0 or 1
    const int l16  = lane & 15;
    const int bk   = half * 2;          // f32 WMMA frag K-offset per half-wave

    const float* adjB = adj + (size_t)b * NNODES * NNODES;
    const float* Yb   = Y   + (size_t)b * NNODES * FD;

    v8f acc = {};

    for (int k0 = 0; k0 < NNODES; k0 += KC) {
        // --- stage A tile: 16 x KC floats (256 float4, 128 threads) ---
        for (int t = tid; t < 16 * (KC / 4); t += 128) {
            const int r  = t >> 4;
            const int c4 = t & 15;
            const float4 v = *(const float4*)(adjB + (size_t)(i0 + r) * NNODES + k0 + c4 * 4);
            float* dst = &ldsA[r * LDSS + c4 * 4];
            ((float2*)dst)[0] = make_float2(v.x, v.y);
            ((float2*)dst)[1] = make_float2(v.z, v.w);
        }
        // --- stage Y tile transposed: ldsY[n][k] (1024 float4, 128 threads) ---
        for (int t = tid; t < KC * (FD / 4); t += 128) {
            const int k  = t >> 4;
            const int ng = t & 15;
            const float4 v = *(const float4*)(Yb + (size_t)(k0 + k) * FD + ng * 4);
            ldsY[(ng * 4 + 0) * LDSS + k] = v.x;
            ldsY[(ng * 4 + 1) * LDSS + k] = v.y;
            ldsY[(ng * 4 + 2) * LDSS + k] = v.z;
            ldsY[(ng * 4 + 3) * LDSS + k] = v.w;
        }
        __syncthreads();

        // prefetch next adj tile (uniform branch; emits global_prefetch_b8)
        if (k0 + KC < NNODES)
            __builtin_prefetch(adjB + (size_t)(i0 + (tid & 15)) * NNODES + k0 + KC + (tid >> 4) * 8, 0, 1);

        const float* arow = &ldsA[l16 * LDSS];
        const float* brow = &ldsY[(wave * 16 + l16) * LDSS];
#pragma unroll
        for (int kk = 0; kk < KC; kk += 4) {
            // f32 WMMA fragment layout:
            //  A: lane(0-15)=row m, v0=A[m,kk+0],v1=A[m,kk+1]; lanes 16-31: K+2
            //  B: lane(0-15)=col n, v0=B[kk+0,n],v1=B[kk+1,n]; lanes 16-31: K+2
            const float2 af = *(const float2*)(arow + kk + bk);
            const float2 bf = *(const float2*)(brow + kk + bk);
            v2f a  = {af.x, af.y};
            v2f bb = {bf.x, bf.y};
            acc = __builtin_amdgcn_wmma_f32_16x16x4_f32(
                false, a, false, bb, (short)0, acc, false, false);
        }
        __syncthreads();
    }

    // epilogue: C/D layout -> row m = v + 8*half, col = wave*16 + l16; apply mask
    const int valid = num_nodes[b];
    const int col   = wave * 16 + l16;
#pragma unroll
    for (int v = 0; v < 8; ++v) {
        const int row = i0 + v + half * 8;
        const float o = (row < valid) ? acc[v] : 0.0f;
        H[((size_t)b * NNODES + row) * FD + col] = o;
    }
}

// ---------------------------------------------------------------------------
// Per-feature sums: stat[f] = sum h, stat[64+f] = sum h^2, stat[128] = total N
// (padded rows of H are already zero -> contribute nothing)
// ---------------------------------------------------------------------------
__global__ __launch_bounds__(256) void stats_kernel(
    const float* __restrict__ H, const int* __restrict__ num_nodes,
    float* __restrict__ stat)
{
    const int f = blockIdx.x;    // one feature per block
    const int tid = threadIdx.x;
    float s = 0.f, q = 0.f;
    for (int i = tid; i < BB * NNODES; i += 256) {
        const float v = H[(size_t)i * FD + f];
        s += v;
        q += v * v;
    }
    __shared__ float rs[256], rq[256];
    rs[tid] = s; rq[tid] = q;
    __syncthreads();
    for (int off = 128; off > 0; off >>= 1) {
        if (tid < off) { rs[tid] += rs[tid + off]; rq[tid] += rq[tid + off]; }
        __syncthreads();
    }
    if (tid == 0) { stat[f] = rs[0]; stat[FD + f] = rq[0]; }
    if (f == 0 && tid == 0) {
        int t = 0;
        for (int bidx = 0; bidx < BB; ++bidx) t += num_nodes[bidx];
        stat[2 * FD] = (float)t;
    }
}

// ---------------------------------------------------------------------------
// out = (h - mean) / std ;  std==0 -> 1e-15
// ---------------------------------------------------------------------------
__global__ __launch_bounds__(256) void norm_kernel(
    const float* __restrict__ H, const float* __restrict__ stat,
    float* __restrict__ out)
{
    const size_t idx = (size_t)blockIdx.x * 256 + threadIdx.x;
    if (idx >= (size_t)BB * NNODES * FD) return;
    const int f = (int)(idx & 63);
    const float total = stat[2 * FD];
    const float mean  = stat[f] / total;
    float var = (stat[FD + f] - total * mean * mean) / (total - 1.0f);
    var = fmaxf(var, 0.0f);
    float sd = sqrtf(var);
    if (sd == 0.0f) sd = 1e-15f;
    out[idx] = (H[idx] - mean) / sd;
}

// ---------------------------------------------------------------------------
extern "C" void kernel_launch(void* const* d_in, const int* in_sizes, int n_in,
                              void* d_out, int out_size, void* d_ws, size_t ws_size,
                              hipStream_t stream)
{
    const float* x    = (const float*)d_in[0];  // [B,N,64]
    const float* adj  = (const float*)d_in[1];  // [B,N,N]
    const int*   nn   = (const int*)  d_in[2];  // [B]
    const float* W1   = (const float*)d_in[3];
    const float* b1   = (const float*)d_in[4];
    const float* W2   = (const float*)d_in[5];
    const float* b2   = (const float*)d_in[6];
    const float* W3   = (const float*)d_in[7];
    const float* b3   = (const float*)d_in[8];
    float* out = (float*)d_out;

    const size_t nElem = (size_t)BB * NNODES * FD;  // 2M floats = 8 MB
    float* Ybuf = (float*)d_ws;           // X@W+b scratch
    float* hA   = Ybuf + nElem;           // layer output ping
    float* hB   = hA + nElem;             // layer output pong
    float* stat = hB + nElem;             // 129 floats of stats

    const dim3 aggGrid(NNODES / 16, BB);
    const int linBlocks = BB * NNODES / 4;

    // layer 1
    linear_kernel<<<linBlocks, 256, 0, stream>>>(x, W1, b1, Ybuf);
    agg_wmma_kernel<<<aggGrid, 128, 0, stream>>>(adj, Ybuf, nn, hA);
    // layer 2
    linear_kernel<<<linBlocks, 256, 0, stream>>>(hA, W2, b2, Ybuf);
    agg_wmma_kernel<<<aggGrid, 128, 0, stream>>>(adj, Ybuf, nn, hB);
    // layer 3
    linear_kernel<<<linBlocks, 256, 0, stream>>>(hB, W3, b3, Ybuf);
    agg_wmma_kernel<<<aggGrid, 128, 0, stream>>>(adj, Ybuf, nn, hA);
    // masked batch-norm
    stats_kernel<<<FD, 256, 0, stream>>>(hA, nn, stat);
    norm_kernel<<<(unsigned)((nElem + 255) / 256), 256, 0, stream>>>(hA, stat, out);
}